// DWHT_1726576853031
// MI455X (gfx1250) — compile-verified
//
#include <hip/hip_runtime.h>
#include <hip/hip_bf16.h>

typedef float v2f __attribute__((ext_vector_type(2)));
typedef float v8f __attribute__((ext_vector_type(8)));

#define BATCH 64
#define CIN   256
#define COUT  512
#define HW    784   // 28*28
#define NPASS 8

// ---------------------------------------------------------------------------
// Kernel 1: build the 512x256 transform matrix T (row-major, ld = 256).
// Column j of T = (shuffle o butterfly^8 o pad)(e_j).
// Per-pass recurrence (o = old state, n = new state), derived from the
// reference's partially-in-place update:
//   n[t]        = o[2t] + o[2t+1]                        t in [0,256)
//   n[256 + i]  = o[4i] + o[4i+1] - o[4i+2] - o[4i+3]    i in [0,128)
//   n[384 + i]  = o[256+2i] - o[257+2i]                  i in [0,128)
// Final channel shuffle (groups=8): out[r] = state[(r%8)*64 + r/8].
// ---------------------------------------------------------------------------
__global__ __launch_bounds__(COUT) void dwht_build_T(float* __restrict__ T) {
  __shared__ float buf[2][COUT];
  const int j = blockIdx.x;
  const int t = threadIdx.x;

  buf[0][t] = (t == j) ? 1.0f : 0.0f;
  __syncthreads();

  int cur = 0;
  #pragma unroll
  for (int p = 0; p < NPASS; ++p) {
    const float* o = buf[cur];
    float*       n = buf[cur ^ 1];
    float v;
    if (t < 256) {
      v = o[2 * t] + o[2 * t + 1];
    } else if (t < 384) {
      const int i = t - 256;
      v = (o[4 * i] + o[4 * i + 1]) - (o[4 * i + 2] + o[4 * i + 3]);
    } else {
      const int i = t - 384;
      v = o[256 + 2 * i] - o[257 + 2 * i];
    }
    n[t] = v;           // reads hit buf[cur], writes hit buf[cur^1]
    __syncthreads();    // one barrier per pass is sufficient
    cur ^= 1;
  }

  const int src = (t & 7) * 64 + (t >> 3);
  T[(size_t)t * CIN + j] = buf[cur][src];
}

// ---------------------------------------------------------------------------
// Kernel 2: Out[b] (512x784) = T (512x256) x Xb (256x784) via
// V_WMMA_F32_16X16X4_F32, N-blocked 2x per wave.
// Grid: (25 pixel groups of 32, 4 row groups, 64 batches).
// Block = 256 threads = 8 waves. Each wave computes a 16x32 output strip
// (16 out-channels x two 16-pixel subtiles): each A fragment (one
// global_load_b64 per lane, L2-resident T) feeds TWO wmma ops.
// X tile stored as TWO independent 16-wide subtiles Xs[2][256][16] so each
// B fragment is one conflict-free ds_load_2addr_b32 (offset1:16) landing in
// a consecutive VGPR pair consumed directly by the WMMA (no v_mov repack;
// word addresses hit banks m, m+16, m+32, m+48 -> all 64 banks once).
// 784 = 49 tiles -> last group half-valid: loads clamp the column (WMMA
// needs EXEC all-ones), second-subtile stores are guarded.
// ---------------------------------------------------------------------------
__global__ __launch_bounds__(256) void dwht_wmma_gemm(const float* __restrict__ X,
                                                      const float* __restrict__ T,
                                                      float* __restrict__ Out) {
  __shared__ float Xs[2][CIN][16];  // two 16-pixel subtiles, 32 KB total

  const int b    = blockIdx.z;
  const int col0 = blockIdx.x * 32;          // first pixel of this group
  const int tid  = threadIdx.x;
  const int lane = tid & 31;
  const int wave = tid >> 5;

  // --- cooperative load of the 256x32 X tile into the two LDS subtiles
  //     (32 consecutive threads fetch 32 consecutive floats of one channel;
  //      clamp col for the tail group)
  {
    const float* Xb = X + (size_t)b * CIN * HW;
    const int p  = tid & 15;                 // pixel within subtile
    const int s  = (tid >> 4) & 1;           // subtile 0/1
    const int c0 = tid >> 5;                 // 0..7
    const int col = col0 + s * 16 + p;
    const int gc  = (col < HW) ? col : (HW - 1);
    #pragma unroll
    for (int it = 0; it < 32; ++it) {
      const int c = it * 8 + c0;             // channel (k index)
      Xs[s][c][p] = Xb[(size_t)c * HW + gc];
    }
  }
  __syncthreads();

  // --- per-wave WMMA strip
  const int row0 = (blockIdx.y * 8 + wave) * 16;  // first out-channel
  const int m    = lane & 15;                     // row (A) / col (B) in tile
  const int kh   = (lane >> 4) << 1;              // K sub-pair: 0 or 2

  // A fragment source: T[row0+m][k+kh], T[row0+m][k+kh+1]  (8B aligned)
  const float* Trow = T + (size_t)(row0 + m) * CIN + kh;

  v8f acc0 = {0.f, 0.f, 0.f, 0.f, 0.f, 0.f, 0.f, 0.f};
  v8f acc1 = {0.f, 0.f, 0.f, 0.f, 0.f, 0.f, 0.f, 0.f};

  #pragma unroll 4
  for (int k = 0; k < CIN; k += 4) {
    v2f a = *(const v2f*)(Trow + k);              // A: 16x4 f32, reused 2x
    v2f b0, b1;                                   // B: 4x16 f32 per subtile
    b0.x = Xs[0][k + kh][m];
    b0.y = Xs[0][k + kh + 1][m];
    b1.x = Xs[1][k + kh][m];
    b1.y = Xs[1][k + kh + 1][m];
    acc0 = __builtin_amdgcn_wmma_f32_16x16x4_f32(false, a, false, b0,
                                                 (short)0, acc0, false, false);
    acc1 = __builtin_amdgcn_wmma_f32_16x16x4_f32(false, a, false, b1,
                                                 (short)0, acc1, false, false);
  }

  // --- store D: VGPR i holds row (i + 8*(lane>=16)), col = lane&15
  const int rbase = row0 + ((lane >> 4) << 3);
  const int c0s   = col0 + (lane & 15);           // subtile 0: always valid
  const int c1s   = c0s + 16;                     // subtile 1: tail-guarded
  float* Ob = Out + (size_t)b * COUT * HW;
  #pragma unroll
  for (int i = 0; i < 8; ++i) {
    Ob[(size_t)(rbase + i) * HW + c0s] = acc0[i];
  }
  if (col0 + 16 < HW) {
    #pragma unroll
    for (int i = 0; i < 8; ++i) {
      Ob[(size_t)(rbase + i) * HW + c1s] = acc1[i];
    }
  }
}

// ---------------------------------------------------------------------------
extern "C" void kernel_launch(void* const* d_in, const int* in_sizes, int n_in,
                              void* d_out, int out_size, void* d_ws, size_t ws_size,
                              hipStream_t stream) {
  (void)in_sizes; (void)n_in; (void)out_size; (void)ws_size;
  const float* x   = (const float*)d_in[0];     // (64, 256, 28, 28) f32
  float*       out = (float*)d_out;             // (64, 512, 28, 28) f32
  float*       T   = (float*)d_ws;              // 512*256*4 = 512 KB scratch

  dwht_build_T<<<dim3(CIN), dim3(COUT), 0, stream>>>(T);

  dim3 grid((HW + 31) / 32, COUT / (16 * 8), BATCH);   // (25, 4, 64)
  dwht_wmma_gemm<<<grid, dim3(256), 0, stream>>>(x, T, out);
}